// Project2Dto3D_89670327206299
// MI455X (gfx1250) — compile-verified
//
#include <hip/hip_runtime.h>
#include <cstdint>

// Project2Dto3D scatter-add for MI455X (gfx1250, wave32).
//
// out[b,c,idx[p]] += feat[b,c,p]   for all p in HW, (b,c) in B*C
//
// Memory-bound: ~157MB feature stream (read once, NT) + 265MB zero-fill (RT,
// kept L2-resident) + 39.3M f32 atomics resolved in L2 (192MB). No matrix
// math -> WMMA inapplicable; CDNA5 features used: async global->LDS pipe
// (ASYNCcnt) for index staging, s_wait_asynccnt, NT temporal hints, native
// no-return global_atomic_add_f32 at AGENT scope.

#define BDIM 256

constexpr int kB  = 4;
constexpr int kC  = 128;
constexpr int kH  = 240;
constexpr int kW  = 320;
constexpr int kHW = kH * kW;            // 76800
constexpr int kV  = 60 * 36 * 60;       // 129600
constexpr int kBC = kB * kC;            // 512
constexpr long long kOutFloat4 = (long long)kBC * kV / 4;   // 16,588,800
constexpr int kZeroBlocks = (int)(kOutFloat4 / BDIM);       // 64,800 exact

// Native vector type so __builtin_nontemporal_load accepts it.
typedef float vfloat4 __attribute__((ext_vector_type(4)));

// ---------------------------------------------------------------------------
// Zero-fill with 16B stores. Regular-temporal so the zeros stay resident in
// L2, where the subsequent atomics are resolved.
// ---------------------------------------------------------------------------
__global__ __launch_bounds__(BDIM) void zero_out_kernel(vfloat4* __restrict__ out)
{
    size_t i = (size_t)blockIdx.x * BDIM + threadIdx.x;
    out[i] = (vfloat4){0.0f, 0.0f, 0.0f, 0.0f};
}

// ---------------------------------------------------------------------------
// Scatter kernel.
//   grid = (BC=512, HW/4/BDIM=75), block = 256 (8 wave32s).
//   blockIdx.x = bc plane -> consecutive blocks reuse the same 8KB index slab
//   (WGP$/L2 resident).
//   Each thread: 4 consecutive pixels of one (b,c) plane.
//     - 32B of int64 indices staged via async global->LDS (2x b128, ASYNCcnt)
//     - 16B coalesced non-temporal float4 feature load (read-once stream;
//       NT keeps L2 free for the 265MB atomic accumulator)
//     - 4x global_atomic_add_f32 (no-return, agent scope, relaxed)
// ---------------------------------------------------------------------------
__global__ __launch_bounds__(BDIM) void scatter_add_kernel(
    const float*     __restrict__ feat,   // [BC, HW]
    const long long* __restrict__ idx64,  // [HW], values in [0, V)
    float*           __restrict__ out)    // [BC, V]
{
    const int tid = threadIdx.x;
    const int bc  = blockIdx.x;                       // 0..511
    const int p0  = (blockIdx.y * BDIM + tid) * 4;    // pixel quad base

    // 32 bytes of index data per thread.
    __shared__ int4 sIdx[BDIM * 2];

    // LDS byte offset for VDST: low 32 bits of the generic pointer are the
    // in-allocation LDS offset (aperture is keyed on ADDR[63:32]).
    const uint32_t lds_off = (uint32_t)(uintptr_t)(&sIdx[tid * 2]);
    const uint64_t gaddr   = (uint64_t)(uintptr_t)(idx64 + p0);

    // gfx1250 async global->LDS copy (GV mode). INST_OFFSET applies to both
    // the memory and LDS sides, so one base covers both 16B halves.
    asm volatile(
        "global_load_async_to_lds_b128 %0, %1, off\n\t"
        "global_load_async_to_lds_b128 %0, %1, off offset:16"
        :
        : "v"(lds_off), "v"(gaddr)
        : "memory");

    // Overlap: stream the feature quad (non-temporal, read-once) while the
    // async index copy is in flight.
    const vfloat4 f = __builtin_nontemporal_load(
        (const vfloat4*)(feat + (size_t)bc * kHW + p0));

    // Wait for this wave's async LDS writes; each lane reads only its own
    // slot, so no workgroup barrier is required.
    asm volatile("s_wait_asynccnt 0" ::: "memory");

    const int4 a = sIdx[tid * 2 + 0];   // idx64[p0]   (a.x=lo), idx64[p0+1] (a.z=lo)
    const int4 b = sIdx[tid * 2 + 1];   // idx64[p0+2] (b.x=lo), idx64[p0+3] (b.z=lo)

    float* __restrict__ op = out + (size_t)bc * kV;
    __hip_atomic_fetch_add(op + a.x, f.x, __ATOMIC_RELAXED, __HIP_MEMORY_SCOPE_AGENT);
    __hip_atomic_fetch_add(op + a.z, f.y, __ATOMIC_RELAXED, __HIP_MEMORY_SCOPE_AGENT);
    __hip_atomic_fetch_add(op + b.x, f.z, __ATOMIC_RELAXED, __HIP_MEMORY_SCOPE_AGENT);
    __hip_atomic_fetch_add(op + b.z, f.w, __ATOMIC_RELAXED, __HIP_MEMORY_SCOPE_AGENT);
}

extern "C" void kernel_launch(void* const* d_in, const int* in_sizes, int n_in,
                              void* d_out, int out_size, void* d_ws, size_t ws_size,
                              hipStream_t stream)
{
    const float*     feat  = (const float*)d_in[0];      // [4,128,240,320] f32
    const long long* idx64 = (const long long*)d_in[1];  // [240,320] i64
    float*           out   = (float*)d_out;              // [4,128,129600] f32

    // 1) Zero the accumulator (must happen every launch; harness replays).
    zero_out_kernel<<<dim3(kZeroBlocks), BDIM, 0, stream>>>((vfloat4*)out);

    // 2) Scatter-add. 512 (b,c) planes x 75 pixel-groups.
    dim3 grid(kBC, kHW / 4 / BDIM);   // (512, 75)
    scatter_add_kernel<<<grid, BDIM, 0, stream>>>(feat, idx64, out);
}